// AntialiasedBicubic_223338299791
// MI455X (gfx1250) — compile-verified
//
#include <hip/hip_runtime.h>
#include <cmath>

// ---------------------------------------------------------------------------
// Fused antialiased 4x bicubic downscale (blur3 -> bicubic/2 -> blur3 ->
// bicubic/2) as a single separable 16-tap stride-4 filter per dimension.
// Single pass over the input, staged through LDS with CDNA5 async global->LDS
// DMA (b128 fast path, b32 clamped path at x-borders). LDS reads vectorized
// (ds_load_b128) and phase-3 tap windows shared across 4 output rows.
// ---------------------------------------------------------------------------

#define TOX 32            // output tile width  (per workgroup)
#define TOY 32            // output tile height (per workgroup)
#define IN_ROWS 140       // 4*TOY + 12 (16-tap stride-4 footprint)
#define LD_COLS 144       // aligned load window: 4*TOX + 16, starts at -8
#define IN_STRIDE 148     // padded LDS row stride (floats, multiple of 4)
#define IMG_H 1024
#define IMG_W 1024
#define OUT_H 256
#define OUT_W 256

struct Filt16 { float f[16]; };

typedef __attribute__((address_space(3))) float lds_float;
typedef __attribute__((ext_vector_type(4))) float v4f;

__global__ __launch_bounds__(256)
void aa_bicubic_down4_kernel(const float* __restrict__ in,
                             float* __restrict__ out,
                             Filt16 Fw)
{
    __shared__ float s_in[IN_ROWS * IN_STRIDE];   // 140*148*4 = 82880 B
    __shared__ float s_mid[IN_ROWS * TOX];        // 140*32*4  = 17920 B

    const int tid = threadIdx.x;
    const int tx  = tid & 31;     // 0..31
    const int ty  = tid >> 5;     // 0..7

    const int tile_x = blockIdx.x;   // 0..7
    const int tile_y = blockIdx.y;   // 0..7
    const int slice  = blockIdx.z;   // 0..N*C-1

    const float* __restrict__ src = in  + (size_t)slice * (IMG_H * IMG_W);
    float*       __restrict__ dst = out + (size_t)slice * (OUT_H * OUT_W);

    // LDS column c holds global column (gx0a + c); output tap k of column cx
    // reads global col tile_x*128 + 4*cx + k - 6 = gx0a + (4*cx + k + 2).
    const int gx0a = tile_x * (4 * TOX) - 8;   // 16B-aligned window start
    const int gy0  = tile_y * (4 * TOY) - 6;

    const bool xborder = (tile_x == 0) || (tile_x == (int)gridDim.x - 1);

    // ---- Phase 1: async DMA global -> LDS ----------------------------------
    if (!xborder) {
        // Fast path: 36 x b128 per row, 7 rows per sweep of 252 threads.
        const int r7 = tid / 36;        // 0..7 (7 valid)
        const int q  = tid % 36;        // qword within row
        if (r7 < 7) {
            const int col = q * 4;      // float column, 16B aligned
            for (int i = 0; i < 20; ++i) {
                const int r = r7 + 7 * i;              // covers 0..139 exactly
                int gy = gy0 + r;
                gy = gy < 0 ? 0 : (gy > IMG_H - 1 ? IMG_H - 1 : gy);
                const float* gp = src + (size_t)gy * IMG_W + (gx0a + col);
                lds_float* lp = (lds_float*)&s_in[r * IN_STRIDE + col];
                asm volatile("global_load_async_to_lds_b128 %0, %1, off"
                             :: "v"(lp), "v"(gp) : "memory");
            }
        }
    } else {
        // Border path: per-element clamped b32 async copies.
        for (int i = 0; i < 18; ++i) {
            const int r = ty + 8 * i;
            if (r >= IN_ROWS) break;
            int gy = gy0 + r;
            gy = gy < 0 ? 0 : (gy > IMG_H - 1 ? IMG_H - 1 : gy);
            const float* grow = src + (size_t)gy * IMG_W;
#pragma unroll
            for (int j = 0; j < 5; ++j) {
                const int c = tx + 32 * j;
                if (c < LD_COLS) {
                    int gx = gx0a + c;
                    gx = gx < 0 ? 0 : (gx > IMG_W - 1 ? IMG_W - 1 : gx);
                    lds_float* lp = (lds_float*)&s_in[r * IN_STRIDE + c];
                    asm volatile("global_load_async_to_lds_b32 %0, %1, off"
                                 :: "v"(lp), "v"(grow + gx) : "memory");
                }
            }
        }
    }
    asm volatile("s_wait_asynccnt 0" ::: "memory");
    __syncthreads();

    // ---- Phase 2: horizontal 16-tap, stride 4 ------------------------------
    // Aligned 20-float window per lane: 5 x ds_load_b128, taps at t[2..17].
    for (int i = 0; i < 18; ++i) {
        const int r = ty + 8 * i;
        if (r >= IN_ROWS) break;
        const v4f* qp = (const v4f*)&s_in[r * IN_STRIDE + 4 * tx];
        float t[20];
#pragma unroll
        for (int j = 0; j < 5; ++j)
            ((v4f*)t)[j] = qp[j];
        float acc = 0.0f;
#pragma unroll
        for (int k = 0; k < 16; ++k)
            acc = fmaf(Fw.f[k], t[2 + k], acc);
        s_mid[r * TOX + tx] = acc;
    }
    __syncthreads();

    // ---- Phase 3: vertical 16-tap, stride 4 + store ------------------------
    // Each thread produces 4 consecutive output rows; their tap windows
    // overlap, so load the union (28 rows) once.
    {
        float t[28];
#pragma unroll
        for (int m = 0; m < 28; ++m)
            t[m] = s_mid[(16 * ty + m) * TOX + tx];

        const int oxBase = tile_x * TOX + tx;
#pragma unroll
        for (int u = 0; u < 4; ++u) {
            float acc = 0.0f;
#pragma unroll
            for (int k = 0; k < 16; ++k)
                acc = fmaf(Fw.f[k], t[4 * u + k], acc);
            const int oy = tile_y * TOY + 4 * ty + u;
            dst[(size_t)oy * OUT_W + oxBase] = acc;
        }
    }
}

extern "C" void kernel_launch(void* const* d_in, const int* in_sizes, int n_in,
                              void* d_out, int out_size, void* d_ws, size_t ws_size,
                              hipStream_t stream)
{
    (void)n_in; (void)out_size; (void)d_ws; (void)ws_size;

    const float* x = (const float*)d_in[0];
    float* y = (float*)d_out;

    // Compose blur(3, sigma=0.66) with the fixed 4-tap bicubic half
    // downsample, applied twice: single 16-tap stride-4 filter.
    Filt16 F;
    {
        const double sigma = 0.66;
        double g[3];
        for (int i = 0; i < 3; ++i) {
            const double t = (double)(i - 1);
            g[i] = std::exp(-0.5 * t * t / (sigma * sigma));
        }
        const double gs = g[0] + g[1] + g[2];
        for (int i = 0; i < 3; ++i) g[i] /= gs;

        const double w[4] = {-0.09375, 0.59375, 0.59375, -0.09375};
        double c[6] = {0, 0, 0, 0, 0, 0};          // one blur+down2 stage
        for (int t = 0; t < 4; ++t)
            for (int u = 0; u < 3; ++u)
                c[t + u] += w[t] * g[u];

        double Fd[16] = {0};                        // two stages composed
        for (int s2 = 0; s2 < 6; ++s2)
            for (int s1 = 0; s1 < 6; ++s1)
                Fd[2 * s2 + s1] += c[s2] * c[s1];
        for (int k = 0; k < 16; ++k) F.f[k] = (float)Fd[k];
    }

    const int slices = in_sizes[0] / (IMG_H * IMG_W);   // N*C = 48
    dim3 grid(IMG_W / (4 * TOX),    // 8
              IMG_H / (4 * TOY),    // 8
              slices);              // 48
    aa_bicubic_down4_kernel<<<grid, dim3(256, 1, 1), 0, stream>>>(x, y, F);
}